// DFTQNN_81776177316168
// MI455X (gfx1250) — compile-verified
//
#include <hip/hip_runtime.h>
#include <hip/hip_bf16.h>

// ---------------------------------------------------------------------------
// DFT-QNN statevector simulator for MI455X (gfx1250, wave32, WMMA).
//
// Algorithm: for each gate k,
//   G = (M + M^H)/2,  A = (-i theta_k / 2^10) * G   (||A|| ~ 0.2)
//   T = TaylorHorner(A, order 6)   -> 5 complex GEMMs (WMMA f32 16x16x4)
//   T = T^2 ten times              -> 10 complex GEMMs
//   psi = T @ psi                  -> complex GEMV (wave-per-row)
// probs = |psi|^2.
//
// Generators stream from HBM once (192 MB); working set (24 MB) is
// L2-resident; the run is f32-WMMA compute bound. Each wave register-blocks
// a 32x32 complex output tile (2x2 WMMA sub-tiles); K-panels are fetched
// straight into LDS with GLOBAL_LOAD_ASYNC_TO_LDS_B128 (ASYNCcnt) and
// double-buffered so the async copy of panel k+1 overlaps the WMMA chain
// on panel k.
// Workspace requirement: 6 * 1024^2 * 4 B + 16 KB ~= 24.02 MB.
// ---------------------------------------------------------------------------

typedef float v2f __attribute__((ext_vector_type(2)));
typedef float v8f __attribute__((ext_vector_type(8)));

#define DIMQ 1024
#define D2Q (1024 * 1024)

// LDS byte offset of a __shared__ object: generic LDS addresses carry the
// wave-relative offset in addr[31:0] (ISA 10.2 aperture mapping).
__device__ __forceinline__ unsigned qnn_lds_off(const void* p) {
  return (unsigned)(uintptr_t)p;
}

// Async global -> LDS copy of 16 bytes (tracked by ASYNCcnt).
__device__ __forceinline__ void qnn_async_b128(unsigned lds, const float* g) {
  asm volatile("global_load_async_to_lds_b128 %0, %1, off"
               :
               : "v"(lds), "v"((unsigned long long)(uintptr_t)g)
               : "memory");
}

__device__ __forceinline__ void qnn_wait_async0() {
  asm volatile("s_wait_asynccnt 0" ::: "memory");
}

// ---------------------------------------------------------------------------
// psi init: L2-normalized amplitude embedding of `feature` (padded w/ zeros).
// One block of 1024 threads.
__global__ void qnn_init_psi(const float* __restrict__ feature, int featLen,
                             float* __restrict__ pr, float* __restrict__ pi) {
  __shared__ float red[1024];
  int i = threadIdx.x;
  float v = (i < featLen) ? feature[i] : 0.0f;
  red[i] = v * v;
  __syncthreads();
  for (int s = 512; s > 0; s >>= 1) {
    if (i < s) red[i] += red[i + s];
    __syncthreads();
  }
  float inv = rsqrtf(red[0]);
  pr[i] = v * inv;
  pi[i] = 0.0f;
}

// ---------------------------------------------------------------------------
// A = (-i * theta / 2^s) * Herm(G):  A_re = c*Gi_anti, A_im = -c*Gr_sym
__global__ void qnn_prep_A(const float* __restrict__ gre,
                           const float* __restrict__ gim,
                           const float* __restrict__ theta, int k, float inv2s,
                           float* __restrict__ Are, float* __restrict__ Aim) {
  int idx = blockIdx.x * blockDim.x + threadIdx.x;
  int i = idx >> 10;
  int j = idx & 1023;
  float c = theta[k] * inv2s;
  float sym_re = 0.5f * (gre[(size_t)i * DIMQ + j] + gre[(size_t)j * DIMQ + i]);
  float sym_im = 0.5f * (gim[(size_t)i * DIMQ + j] - gim[(size_t)j * DIMQ + i]);
  Are[idx] = c * sym_im;
  Aim[idx] = -c * sym_re;
}

// ---------------------------------------------------------------------------
// Horner seed: B = I + A/m
__global__ void qnn_set_B(const float* __restrict__ Are,
                          const float* __restrict__ Aim, float invm,
                          float* __restrict__ Bre, float* __restrict__ Bim) {
  int idx = blockIdx.x * blockDim.x + threadIdx.x;
  int i = idx >> 10;
  int j = idx & 1023;
  Bre[idx] = Are[idx] * invm + ((i == j) ? 1.0f : 0.0f);
  Bim[idx] = Aim[idx] * invm;
}

// ---------------------------------------------------------------------------
// Complex GEMM: C = alpha * (A @ B) + (addIdent ? I : 0), all 1024x1024 f32.
// Block: 128 threads = 4 waves; macro-tile 64x64; each wave register-blocks a
// 32x32 output tile = 2x2 sub-tiles of V_WMMA_F32_16X16X4_F32 (re + im).
// K-panels double-buffered in LDS via GLOBAL_LOAD_ASYNC_TO_LDS_B128.
#define ASTR 36  // A panel LDS row stride (mult of 4 -> 16B aligned B128 dst)
#define BSTR 68  // B panel LDS row stride
__global__ void __launch_bounds__(128)
qnn_cgemm(const float* __restrict__ Ar, const float* __restrict__ Ai,
          const float* __restrict__ Br, const float* __restrict__ Bi,
          float* __restrict__ Cr, float* __restrict__ Ci, float alpha,
          int addIdent) {
  __shared__ float sAr[2][64][ASTR];
  __shared__ float sAi[2][64][ASTR];
  __shared__ float sBr[2][32][BSTR];
  __shared__ float sBi[2][32][BSTR];

  const int t = threadIdx.x;
  const int lane = t & 31;
  const int wave = t >> 5;            // 0..3
  const int row0 = (wave >> 1) * 32;  // wave's 32-row band in macro-tile
  const int col0 = (wave & 1) * 32;   // wave's 32-col band in macro-tile
  const int rowBase = blockIdx.y * 64;
  const int colBase = blockIdx.x * 64;

  const int m = lane & 15;
  const int koff = (lane < 16) ? 0 : 2;

  // Issue async copies of one 32-wide K panel into LDS buffer `buf`.
  auto issuePanel = [&](int buf, int k0) {
    // A panel: 64 rows x 32 cols, 16B vectors -> 512 transfers per matrix.
#pragma unroll
    for (int v = t; v < 512; v += 128) {
      int r = v >> 3, c4 = (v & 7) << 2;
      size_t g = (size_t)(rowBase + r) * DIMQ + (k0 + c4);
      qnn_async_b128(qnn_lds_off(&sAr[buf][r][c4]), Ar + g);
      qnn_async_b128(qnn_lds_off(&sAi[buf][r][c4]), Ai + g);
    }
    // B panel: 32 rows x 64 cols -> 512 transfers per matrix.
#pragma unroll
    for (int v = t; v < 512; v += 128) {
      int r = v >> 4, c4 = (v & 15) << 2;
      size_t g = (size_t)(k0 + r) * DIMQ + (colBase + c4);
      qnn_async_b128(qnn_lds_off(&sBr[buf][r][c4]), Br + g);
      qnn_async_b128(qnn_lds_off(&sBi[buf][r][c4]), Bi + g);
    }
  };

  v8f accR[2][2] = {};
  v8f accI[2][2] = {};

  // Prologue: fetch panel 0.
  issuePanel(0, 0);
  qnn_wait_async0();
  __syncthreads();

  int cur = 0;
  for (int k0 = 0; k0 < DIMQ; k0 += 32) {
    int nxt = cur ^ 1;
    if (k0 + 32 < DIMQ) issuePanel(nxt, k0 + 32);  // overlap with compute

#pragma unroll
    for (int kk = 0; kk < 32; kk += 4) {
      // A fragments (16x4 f32) for two 16-row sub-tiles:
      // lanes 0-15 carry K=kk+0/1, lanes 16-31 carry K=kk+2/3.
      v2f ar[2], ai[2], nai[2];
#pragma unroll
      for (int p = 0; p < 2; ++p) {
        int rr = row0 + p * 16 + m;
        ar[p].x = sAr[cur][rr][kk + koff];
        ar[p].y = sAr[cur][rr][kk + koff + 1];
        ai[p].x = sAi[cur][rr][kk + koff];
        ai[p].y = sAi[cur][rr][kk + koff + 1];
        nai[p] = -ai[p];  // f32 WMMA has no A-neg modifier: negate in VALU
      }
      // B fragments (4x16 f32) for two 16-col sub-tiles:
      // VGPR0 row K=kk+koff, VGPR1 row K=kk+koff+1.
      v2f br[2], bi[2];
#pragma unroll
      for (int q = 0; q < 2; ++q) {
        int cc = col0 + q * 16 + m;
        br[q].x = sBr[cur][kk + koff][cc];
        br[q].y = sBr[cur][kk + koff + 1][cc];
        bi[q].x = sBi[cur][kk + koff][cc];
        bi[q].y = sBi[cur][kk + koff + 1][cc];
      }
      // 16 WMMAs per K-step: (Ar+iAi)(Br+iBi) accumulation
#pragma unroll
      for (int p = 0; p < 2; ++p) {
#pragma unroll
        for (int q = 0; q < 2; ++q) {
          accR[p][q] = __builtin_amdgcn_wmma_f32_16x16x4_f32(
              false, ar[p], false, br[q], (short)0, accR[p][q], false, false);
          accR[p][q] = __builtin_amdgcn_wmma_f32_16x16x4_f32(
              false, nai[p], false, bi[q], (short)0, accR[p][q], false, false);
          accI[p][q] = __builtin_amdgcn_wmma_f32_16x16x4_f32(
              false, ar[p], false, bi[q], (short)0, accI[p][q], false, false);
          accI[p][q] = __builtin_amdgcn_wmma_f32_16x16x4_f32(
              false, ai[p], false, br[q], (short)0, accI[p][q], false, false);
        }
      }
    }

    qnn_wait_async0();  // this wave's prefetch of `nxt` has landed in LDS
    __syncthreads();    // ... and so has every other wave's
    cur = nxt;
  }

  // C/D layout: VGPR r -> M = r (lanes 0-15) or r+8 (lanes 16-31), N = lane&15
  const int moff = (lane < 16) ? 0 : 8;
#pragma unroll
  for (int p = 0; p < 2; ++p) {
#pragma unroll
    for (int q = 0; q < 2; ++q) {
      int gc = colBase + col0 + q * 16 + m;
#pragma unroll
      for (int r = 0; r < 8; ++r) {
        int gr = rowBase + row0 + p * 16 + moff + r;
        float vr = alpha * accR[p][q][r];
        float vi = alpha * accI[p][q][r];
        if (addIdent && gr == gc) vr += 1.0f;
        Cr[(size_t)gr * DIMQ + gc] = vr;
        Ci[(size_t)gr * DIMQ + gc] = vi;
      }
    }
  }
}

// ---------------------------------------------------------------------------
// Complex GEMV: q = T @ p, one wave per row, shfl_xor reduction (wave32).
__global__ void __launch_bounds__(256)
qnn_matvec(const float* __restrict__ Tr, const float* __restrict__ Ti,
           const float* __restrict__ pr, const float* __restrict__ pi,
           float* __restrict__ qr, float* __restrict__ qi) {
  int lane = threadIdx.x & 31;
  int row = blockIdx.x * 8 + (threadIdx.x >> 5);
  float sr = 0.0f, si = 0.0f;
  const float* tr = Tr + (size_t)row * DIMQ;
  const float* ti = Ti + (size_t)row * DIMQ;
  for (int j = lane; j < DIMQ; j += 32) {
    float ar = tr[j], ai = ti[j];
    float xr = pr[j], xi = pi[j];
    sr += ar * xr - ai * xi;
    si += ar * xi + ai * xr;
  }
#pragma unroll
  for (int mask = 16; mask > 0; mask >>= 1) {
    sr += __shfl_xor(sr, mask, 32);
    si += __shfl_xor(si, mask, 32);
  }
  if (lane == 0) {
    qr[row] = sr;
    qi[row] = si;
  }
}

// ---------------------------------------------------------------------------
__global__ void qnn_probs(const float* __restrict__ pr,
                          const float* __restrict__ pi,
                          float* __restrict__ out) {
  int i = blockIdx.x * blockDim.x + threadIdx.x;
  out[i] = pr[i] * pr[i] + pi[i] * pi[i];
}

// ---------------------------------------------------------------------------
extern "C" void kernel_launch(void* const* d_in, const int* in_sizes, int n_in,
                              void* d_out, int out_size, void* d_ws,
                              size_t ws_size, hipStream_t stream) {
  const float* feature = (const float*)d_in[0];  // [1000]
  const float* theta = (const float*)d_in[1];    // [24,1]
  const float* gre = (const float*)d_in[2];      // [24,1024,1024]
  const float* gim = (const float*)d_in[3];      // [24,1024,1024]
  float* out = (float*)d_out;                    // [1024]

  const int featLen = in_sizes[0];
  const int nGates = in_sizes[1];  // 24

  float* ws = (float*)d_ws;
  float* Are = ws + (size_t)0 * D2Q;
  float* Aim = ws + (size_t)1 * D2Q;
  float* Xre = ws + (size_t)2 * D2Q;
  float* Xim = ws + (size_t)3 * D2Q;
  float* Yre = ws + (size_t)4 * D2Q;
  float* Yim = ws + (size_t)5 * D2Q;
  float* vec = ws + (size_t)6 * D2Q;
  float* p0r = vec + 0 * DIMQ;
  float* p0i = vec + 1 * DIMQ;
  float* p1r = vec + 2 * DIMQ;
  float* p1i = vec + 3 * DIMQ;

  qnn_init_psi<<<1, 1024, 0, stream>>>(feature, featLen, p0r, p0i);

  const int S = 10;  // squarings: scale by 2^-10
  const int M = 6;   // Taylor order
  const float inv2s = 1.0f / 1024.0f;

  dim3 gemmGrid(DIMQ / 64, DIMQ / 64);

  float* pr = p0r;
  float* pi_ = p0i;
  float* qr = p1r;
  float* qi = p1i;

  for (int k = 0; k < nGates; ++k) {
    qnn_prep_A<<<D2Q / 256, 256, 0, stream>>>(
        gre + (size_t)k * D2Q, gim + (size_t)k * D2Q, theta, k, inv2s, Are,
        Aim);

    float* Br_ = Xre;
    float* Bi_ = Xim;
    float* Cr_ = Yre;
    float* Ci_ = Yim;
    qnn_set_B<<<D2Q / 256, 256, 0, stream>>>(Are, Aim, 1.0f / (float)M, Br_,
                                             Bi_);
    // Horner: B <- I + (A/j) * B, j = M-1 .. 1
    for (int j = M - 1; j >= 1; --j) {
      qnn_cgemm<<<gemmGrid, 128, 0, stream>>>(Are, Aim, Br_, Bi_, Cr_, Ci_,
                                              1.0f / (float)j, 1);
      float* t;
      t = Br_; Br_ = Cr_; Cr_ = t;
      t = Bi_; Bi_ = Ci_; Ci_ = t;
    }
    // Squaring: T <- T * T, S times
    for (int sq = 0; sq < S; ++sq) {
      qnn_cgemm<<<gemmGrid, 128, 0, stream>>>(Br_, Bi_, Br_, Bi_, Cr_, Ci_,
                                              1.0f, 0);
      float* t;
      t = Br_; Br_ = Cr_; Cr_ = t;
      t = Bi_; Bi_ = Ci_; Ci_ = t;
    }
    // psi <- T @ psi
    qnn_matvec<<<DIMQ / 8, 256, 0, stream>>>(Br_, Bi_, pr, pi_, qr, qi);
    float* t;
    t = pr; pr = qr; qr = t;
    t = pi_; pi_ = qi; qi = t;
  }

  qnn_probs<<<DIMQ / 256, 256, 0, stream>>>(pr, pi_, out);
}